// GraphConvDecoder_45320494907734
// MI455X (gfx1250) — compile-verified
//
#include <hip/hip_runtime.h>

#define N_NODES   100000
#define N_EDGES   1200000
#define NUM_GRAPHS 128
#define IN_DIM    5
#define HIDDEN    64

typedef __attribute__((ext_vector_type(2))) float v2f;
typedef __attribute__((ext_vector_type(8))) float v8f;

// ---------------------------------------------------------------- utilities
__global__ void zero_kernel(float* __restrict__ p, int n) {
    int i = blockIdx.x * blockDim.x + threadIdx.x;
    if (i < n) p[i] = 0.0f;
}

// ------------------------------------------------- layer 1: edge scatter-add
__global__ void scatter1_kernel(const float* __restrict__ x,
                                const int* __restrict__ ei,
                                float* __restrict__ agg1) {
    int e = blockIdx.x * blockDim.x + threadIdx.x;
    if (e >= N_EDGES) return;
    int src = ei[e];
    int dst = ei[N_EDGES + e];
    const float* s = x + (long)src * IN_DIM;
    float* d = agg1 + (long)dst * IN_DIM;
#pragma unroll
    for (int k = 0; k < IN_DIM; ++k) atomicAdd(&d[k], s[k]);
}

// -------------------------------- layer 1 dense: h1 = relu(agg@Wr + b + x@W0)
__global__ void layer1_kernel(const float* __restrict__ x,
                              const float* __restrict__ agg1,
                              const float* __restrict__ w_rel1,
                              const float* __restrict__ b_rel1,
                              const float* __restrict__ w_root1,
                              float* __restrict__ h1) {
    long tid = (long)blockIdx.x * blockDim.x + threadIdx.x;
    if (tid >= (long)N_NODES * HIDDEN) return;
    int n = (int)(tid >> 6);
    int j = (int)(tid & 63);
    float acc = b_rel1[j];
#pragma unroll
    for (int k = 0; k < IN_DIM; ++k) {
        acc += agg1[(long)n * IN_DIM + k] * w_rel1[k * HIDDEN + j];
        acc += x[(long)n * IN_DIM + k]    * w_root1[k * HIDDEN + j];
    }
    h1[tid] = fmaxf(acc, 0.0f);
}

// ------------------------------------------------- layer 2: edge scatter-add
// 8 threads per edge, 8 contiguous floats per thread
__global__ void scatter2_kernel(const float* __restrict__ h1,
                                const int* __restrict__ ei,
                                float* __restrict__ agg2) {
    long tid = (long)blockIdx.x * blockDim.x + threadIdx.x;
    if (tid >= (long)N_EDGES * 8) return;
    int e    = (int)(tid >> 3);
    int base = (int)(tid & 7) * 8;
    int src = ei[e];
    int dst = ei[N_EDGES + e];
    const float* s = h1 + (long)src * HIDDEN + base;
    float* d = agg2 + (long)dst * HIDDEN + base;
#pragma unroll
    for (int i = 0; i < 8; ++i) atomicAdd(&d[i], s[i]);
}

// ---- layer 2 dense via WMMA f32 16x16x4, fused bias+relu+graph add-pooling
// One wave computes one 16(M-rows)x16(N-cols) tile of
//   h2 = relu(agg2 @ w_rel2 + b_rel2 + h1 @ w_root2)
// and atomically pools it into pool_add[batch[row]][col].
__global__ void __launch_bounds__(128)
layer2_pool_kernel(const float* __restrict__ agg2,
                   const float* __restrict__ h1,
                   const float* __restrict__ w_rel2,
                   const float* __restrict__ b_rel2,
                   const float* __restrict__ w_root2,
                   const int* __restrict__ batch,
                   float* __restrict__ pool_add) {
    __shared__ float wrel[HIDDEN * HIDDEN];
    __shared__ float wroot[HIDDEN * HIDDEN];
    int tid = threadIdx.x;
    for (int i = tid; i < HIDDEN * HIDDEN; i += 128) {
        wrel[i]  = w_rel2[i];
        wroot[i] = w_root2[i];
    }
    __syncthreads();

    int wave = tid >> 5;          // 4 waves -> 4 N-blocks of 16 cols
    int lane = tid & 31;
    int hi   = lane >> 4;         // lane half selects K sub-pair (ISA layout)
    int l15  = lane & 15;
    int m0   = blockIdx.x * 16;   // 16 node rows per block
    int n0   = wave * 16;

    // clamp row for A loads (tail tile); results for OOB rows are discarded
    int row_a = m0 + l15;
    if (row_a >= N_NODES) row_a = N_NODES - 1;
    const float* a1 = agg2 + (long)row_a * HIDDEN;
    const float* a2 = h1   + (long)row_a * HIDDEN;

    v8f c = {};
    // agg2 @ w_rel2   (K = 64 as 16 steps of K=4)
#pragma unroll
    for (int kk = 0; kk < 16; ++kk) {
        int k0 = kk * 4 + 2 * hi;                 // A: lane half -> K=2*hi,2*hi+1
        v2f a = { a1[k0], a1[k0 + 1] };
        v2f b = { wrel[(k0 + 0) * HIDDEN + n0 + l15],
                  wrel[(k0 + 1) * HIDDEN + n0 + l15] };
        c = __builtin_amdgcn_wmma_f32_16x16x4_f32(false, a, false, b,
                                                  (short)0, c, false, false);
    }
    // + h1 @ w_root2 accumulated into the same C
#pragma unroll
    for (int kk = 0; kk < 16; ++kk) {
        int k0 = kk * 4 + 2 * hi;
        v2f a = { a2[k0], a2[k0 + 1] };
        v2f b = { wroot[(k0 + 0) * HIDDEN + n0 + l15],
                  wroot[(k0 + 1) * HIDDEN + n0 + l15] };
        c = __builtin_amdgcn_wmma_f32_16x16x4_f32(false, a, false, b,
                                                  (short)0, c, false, false);
    }

    float bias = b_rel2[n0 + l15];
    // C/D layout: VGPR r, lanes 0-15 -> M=r, lanes 16-31 -> M=r+8; N=lane&15
#pragma unroll
    for (int r = 0; r < 8; ++r) {
        int row = m0 + r + hi * 8;
        if (row < N_NODES) {
            float v = fmaxf(c[r] + bias, 0.0f);
            atomicAdd(&pool_add[(long)batch[row] * HIDDEN + n0 + l15], v);
        }
    }
}

// ------------------------------------------------------------ graph counts
__global__ void counts_kernel(const int* __restrict__ batch,
                              float* __restrict__ counts) {
    int n = blockIdx.x * blockDim.x + threadIdx.x;
    if (n < N_NODES) atomicAdd(&counts[batch[n]], 1.0f);
}

// --------------------------- final MLP: out = relu(g@Wh1+b)@Wh2+b2, g=[mean,add]
__global__ void mlp_kernel(const float* __restrict__ pool_add,
                           const float* __restrict__ counts,
                           const float* __restrict__ w_h1,
                           const float* __restrict__ b_h1,
                           const float* __restrict__ w_h2,
                           const float* __restrict__ b_h2,
                           float* __restrict__ out) {
    int t = threadIdx.x;               // one thread per graph, single block
    if (t >= NUM_GRAPHS) return;
    float inv = 1.0f / fmaxf(counts[t], 1.0f);
    float addv[HIDDEN];
#pragma unroll
    for (int k = 0; k < HIDDEN; ++k) addv[k] = pool_add[t * HIDDEN + k];
    float sum = 0.0f;
    for (int j = 0; j < HIDDEN; ++j) {
        float acc = b_h1[j];
        // g[k] = mean (k<64) and add (k>=64); fold: add[k]*(inv*W[k][j] + W[k+64][j])
#pragma unroll 8
        for (int k = 0; k < HIDDEN; ++k) {
            float w = inv * w_h1[k * HIDDEN + j] + w_h1[(k + HIDDEN) * HIDDEN + j];
            acc += addv[k] * w;
        }
        acc = fmaxf(acc, 0.0f);
        sum += acc * w_h2[j];
    }
    out[t] = sum + b_h2[0];
}

// ---------------------------------------------------------------- launcher
extern "C" void kernel_launch(void* const* d_in, const int* in_sizes, int n_in,
                              void* d_out, int out_size, void* d_ws, size_t ws_size,
                              hipStream_t stream) {
    const float* x       = (const float*)d_in[0];
    const int*   ei      = (const int*)  d_in[1];
    const int*   batch   = (const int*)  d_in[2];
    const float* w_rel1  = (const float*)d_in[3];
    const float* b_rel1  = (const float*)d_in[4];
    const float* w_root1 = (const float*)d_in[5];
    const float* w_rel2  = (const float*)d_in[6];
    const float* b_rel2  = (const float*)d_in[7];
    const float* w_root2 = (const float*)d_in[8];
    const float* w_h1    = (const float*)d_in[9];
    const float* b_h1    = (const float*)d_in[10];
    const float* w_h2    = (const float*)d_in[11];
    const float* b_h2    = (const float*)d_in[12];
    float* out = (float*)d_out;

    // workspace carve-out (256B aligned slices)
    char* ws = (char*)d_ws;
    size_t off = 0;
    auto carve = [&](size_t bytes) -> float* {
        float* p = (float*)(ws + off);
        off = (off + bytes + 255) & ~(size_t)255;
        return p;
    };
    float* agg1     = carve((size_t)N_NODES * IN_DIM * sizeof(float));   // 2.0 MB
    float* h1       = carve((size_t)N_NODES * HIDDEN * sizeof(float));   // 25.6 MB
    float* agg2     = carve((size_t)N_NODES * HIDDEN * sizeof(float));   // 25.6 MB
    float* pool_add = carve((size_t)NUM_GRAPHS * HIDDEN * sizeof(float));// 32 KB
    float* counts   = carve((size_t)NUM_GRAPHS * sizeof(float));         // 512 B

    const int B = 256;
    // zero accumulators (required every call; harness does not re-poison)
    zero_kernel<<<(N_NODES * IN_DIM + B - 1) / B, B, 0, stream>>>(agg1, N_NODES * IN_DIM);
    zero_kernel<<<(N_NODES * HIDDEN + B - 1) / B, B, 0, stream>>>(agg2, N_NODES * HIDDEN);
    zero_kernel<<<(NUM_GRAPHS * HIDDEN + B - 1) / B, B, 0, stream>>>(pool_add, NUM_GRAPHS * HIDDEN);
    zero_kernel<<<1, 128, 0, stream>>>(counts, NUM_GRAPHS);

    // layer 1
    scatter1_kernel<<<(N_EDGES + B - 1) / B, B, 0, stream>>>(x, ei, agg1);
    layer1_kernel<<<((long)N_NODES * HIDDEN + B - 1) / B, B, 0, stream>>>(
        x, agg1, w_rel1, b_rel1, w_root1, h1);

    // layer 2
    scatter2_kernel<<<((long)N_EDGES * 8 + B - 1) / B, B, 0, stream>>>(h1, ei, agg2);
    layer2_pool_kernel<<<(N_NODES + 15) / 16, 128, 0, stream>>>(
        agg2, h1, w_rel2, b_rel2, w_root2, batch, pool_add);

    // pooling denominators + readout MLP
    counts_kernel<<<(N_NODES + B - 1) / B, B, 0, stream>>>(batch, counts);
    mlp_kernel<<<1, 128, 0, stream>>>(pool_add, counts, w_h1, b_h1, w_h2, b_h2, out);
}